// FLIF_26130581028962
// MI455X (gfx1250) — compile-verified
//
#include <hip/hip_runtime.h>
#include <math.h>

// FLIF forward on gfx1250.
// - Time blocked into 16-step tiles.
// - Far-field power-law memory term = Toeplitz matmul via V_WMMA_F32_16X16X4_F32
//   (fp32 precision preserved; deltas are O(1e-2) riding on V ~ -70).
// - Near field (<16 lags) + sequential V recurrence per-lane.
// - Input currents staged to LDS with GLOBAL_LOAD_ASYNC_TO_LDS_B32 (ASYNCcnt),
//   double-buffered so HBM latency overlaps the WMMA phase.
// One wave (32 lanes) = 32 neurons = 2 WMMA accumulator tiles; all phases
// wave-private (single-wave workgroups; barriers lower to NOP).

#define TSTEPS 256
#define BS     32768            // B*S = 32*1024
#define TILE   16
#define KCH    4
#define WGT    32               // threads per block (exactly one wave)
#define NEUR   32               // neurons per block

typedef float v2f __attribute__((ext_vector_type(2)));
typedef float v8f __attribute__((ext_vector_type(8)));

__device__ __forceinline__ unsigned lds_offset(const void* p) {
    // generic -> LDS(AS3) pointer, then ptrtoint: 32-bit LDS byte address
    return (unsigned)(size_t)(__attribute__((address_space(3))) const void*)p;
}

__device__ __forceinline__ void async_load_f32_to_lds(unsigned lds_byte_off,
                                                      const float* gptr) {
    unsigned long long ga = (unsigned long long)(size_t)gptr;
    asm volatile("global_load_async_to_lds_b32 %0, %1, off"
                 :: "v"(lds_byte_off), "v"(ga) : "memory");
}

__launch_bounds__(WGT)
__global__ void flif_forward_kernel(const float* __restrict__ I,
                                    const float* __restrict__ V0,
                                    float* __restrict__ out)
{
    __shared__ float wt[TSTEPS + 2];       // power-law kernel, indexed by lag
    __shared__ float dl[TSTEPS][NEUR];     // dl[m][nn] = delta_m  (delta_0 = 0)
    __shared__ float memtile[TILE][NEUR];  // far-field memory for current tile
    __shared__ float Ibuf[2][TILE][NEUR];  // async-staged input currents

    const int tid  = threadIdx.x;                  // == lane id (one wave)
    const int gnid = blockIdx.x * NEUR + tid;      // global neuron id

    // ---- power-law weight table: w[L] = (L+2)^p - (L+1)^p, p = 1 - alpha
    const float p = 0.8f;
    for (int i = tid; i <= TSTEPS; i += WGT)
        wt[i] = powf((float)(i + 2), p) - powf((float)(i + 1), p);
    __syncthreads();   // single-wave WG: lowers to NOP; LDS is in-order per wave

    // ---- stage tile 0 currents (16 outstanding async loads)
    {
        const unsigned l0 = lds_offset(&Ibuf[0][0][tid]);
        const float*   g  = I + gnid;
        #pragma unroll
        for (int s = 0; s < TILE; ++s)
            async_load_f32_to_lds(l0 + s * (NEUR * 4), g + (size_t)s * BS);
    }

    const float THR = -65.0f, V_INIT = -70.0f, VL = -70.0f, GL = 0.025f;
    const float DT_over_tau = 0.005f;                            // DT/(CM/GL)
    const float coef = powf(0.1f, 0.2f) * tgammaf(1.8f) * 2.0f;  // DT^a*G(2-a)/CM
    const float inv_GL = 40.0f;

    float V = V0[gnid];

    const bool hi = (tid >= 16);
    const int  ls = tid & 15;
    const int  ka  = hi ? 2 : 0;         // A-matrix K pair for this lane group
    const int  kb0 = hi ? 2 : 0;         // B-matrix K rows for this lane group
    const int  kb1 = hi ? 3 : 1;
    const int  mrow = hi ? 8 : 0;        // C/D M-offset for this lane group

    for (int tb = 0; tb < TSTEPS / TILE; ++tb) {
        const int n0 = tb * TILE;

        // ---- far field: memtile[i][nn] = sum_{m < n0} wt[(n0+i)-m] * dl[m][nn]
        // A = 16x4 Toeplitz weight block (shared by both neuron groups),
        // B = 4x16 delta block per group, C accumulates 16x16 in fp32.
        v8f c0 = {};
        v8f c1 = {};
        for (int m0 = 0; m0 < n0; m0 += KCH) {
            const int lag0 = (n0 + ls) - (m0 + ka);
            v2f a;  a.x = wt[lag0];  a.y = wt[lag0 - 1];
            v2f b0; b0.x = dl[m0 + kb0][ls];       b0.y = dl[m0 + kb1][ls];
            v2f b1; b1.x = dl[m0 + kb0][16 + ls];  b1.y = dl[m0 + kb1][16 + ls];
            c0 = __builtin_amdgcn_wmma_f32_16x16x4_f32(
                    false, a, false, b0, (short)0, c0, false, false);
            c1 = __builtin_amdgcn_wmma_f32_16x16x4_f32(
                    false, a, false, b1, (short)0, c1, false, false);
        }
        #pragma unroll
        for (int v = 0; v < 8; ++v) {
            memtile[mrow + v][ls]      = c0[v];
            memtile[mrow + v][16 + ls] = c1[v];
        }

        // ---- issue next tile's async stage, then wait for this tile's data.
        // Async loads complete in order: after issuing 16 more, waiting down to
        // <=16 outstanding guarantees the current tile's 16 have landed in LDS.
        if (tb + 1 < TSTEPS / TILE) {
            const unsigned l0 = lds_offset(&Ibuf[(tb + 1) & 1][0][tid]);
            const float*   g  = I + (size_t)(n0 + TILE) * BS + gnid;
            #pragma unroll
            for (int s = 0; s < TILE; ++s)
                async_load_f32_to_lds(l0 + s * (NEUR * 4), g + (size_t)s * BS);
            asm volatile("s_wait_asynccnt 0x10" ::: "memory");
        } else {
            asm volatile("s_wait_asynccnt 0x0" ::: "memory");
        }
        __syncthreads();

        // ---- sequential 16 steps, one neuron per lane
        const float (*Ic)[NEUR] = Ibuf[tb & 1];
        for (int s = 0; s < TILE; ++s) {
            const int n = n0 + s;
            const float It = Ic[s][tid];
            const float spike = (V - THR > 0.0f) ? 1.0f : 0.0f;
            float mem = memtile[s][tid];
            for (int m = n0; m < n; ++m)           // near field: within-tile lags
                mem += wt[n - m] * dl[m][tid];
            float Vn;
            if (n == 0)      Vn = V_INIT;
            else if (n == 1) Vn = V + DT_over_tau * (-V + It * inv_GL);
            else             Vn = coef * (-GL * (V - VL) + It) + V - mem;
            const float spk = (n == 0) ? 0.0f : spike;
            dl[n][tid] = (n >= 1) ? (Vn - V) : 0.0f;   // n==0 zeroes row 0
            const float Vout = Vn - spk * 5.0f;        // THR - V_RESET = 5
            out[(size_t)n * BS + gnid]                       = spk;
            out[(size_t)TSTEPS * BS + (size_t)n * BS + gnid] = Vout;
            V = Vout;
        }
        __syncthreads();
    }
}

extern "C" void kernel_launch(void* const* d_in, const int* in_sizes, int n_in,
                              void* d_out, int out_size, void* d_ws, size_t ws_size,
                              hipStream_t stream) {
    (void)in_sizes; (void)n_in; (void)d_ws; (void)ws_size; (void)out_size;
    const float* I  = (const float*)d_in[0];   // [256, 32, 1024] f32
    const float* V0 = (const float*)d_in[1];   // [32, 1024] f32
    float* out = (float*)d_out;                // [2, 256, 32, 1024] f32
    dim3 grid(BS / NEUR);                      // 1024 blocks
    dim3 block(WGT);                           // 32 threads = 1 wave
    flif_forward_kernel<<<grid, block, 0, stream>>>(I, V0, out);
}